// DMTSkeletonize_13417477832818
// MI455X (gfx1250) — compile-verified
//
#include <hip/hip_runtime.h>
#include <stdint.h>

// DMT skeletonization: exact separable squared-EDT (min-plus, brute force) +
// 26-neighborhood ridge detection. VALU-bound (~8G min-plus steps); working
// set (65.5MB/pass) is L2-resident on MI455X (192MB L2). WMMA is not
// applicable (tropical semiring), so the CDNA5-specific paths used are the
// Tensor Data Mover (tensor_load_to_lds) for strided tile staging and
// global_load_async_to_lds_b32 for contiguous line staging.
//
// Min-plus inner loop uses the exact rearrangement
//   f[j] + (i-j)^2 = (f[j] + j^2) - 2*i*j + i^2
// with w[j] = f[j]+j^2 folded into LDS once per tile, so the per-candidate
// cost is 1 v_fma + 1 v_min per output (all intermediates are integers < 2^24,
// so the fp32 rearrangement is exact for finite values).

#define NN 160
#define NN2 (NN * NN)
#define NN3 (NN * NN * NN)
#define EDT_INF 1000000000000.0f
#define ACC_INIT 3.0e38f

typedef uint32_t u32x4 __attribute__((ext_vector_type(4)));
typedef uint32_t u32x8 __attribute__((ext_vector_type(8)));

// Low 32 bits of a generic pointer to LDS == wave-relative LDS byte address
// (ISA 10.2: LDS aperture uses addr[31:0] as the LDS address).
static __device__ __forceinline__ uint32_t lds_byte_addr(const void* p) {
  return (uint32_t)(uintptr_t)p;
}

// ---------------------------------------------------------------------------
// Pass 1: EDT along W (contiguous). Fused binarize. Block = (160, 2).
// Each block handles 2 lines; lines staged into LDS via async global->LDS.
// ---------------------------------------------------------------------------
__global__ __launch_bounds__(320) void edt_pass_w(const float* __restrict__ img,
                                                  float* __restrict__ dst) {
  __shared__ __align__(16) float line[2][NN];
  const int i = threadIdx.x;                    // 0..159 : output index in line
  const int l = threadIdx.y;                    // 0..1   : which line
  const int y = blockIdx.x * 2 + l;
  const int z = blockIdx.y;
  const int b = blockIdx.z;
  const size_t base = ((size_t)(b * NN + z) * NN + y) * NN;

  // Async copy one f32 per lane into LDS (ASYNCcnt path).
  {
    const float* gp = img + base + i;
    uint32_t loff = lds_byte_addr(&line[l][i]);
    asm volatile("global_load_async_to_lds_b32 %0, %1, off"
                 :: "v"(loff), "v"(gp) : "memory");
    asm volatile("s_wait_asynccnt 0x0" ::: "memory");
  }
  __syncthreads();

  const float fi = (float)i;
  // w[i] = f + i^2, with f = (img > 0.5) ? 0 : INF
  line[l][i] = ((line[l][i] > 0.5f) ? 0.0f : EDT_INF) + fi * fi;
  __syncthreads();

  const float ci = -2.0f * fi;
  float acc = ACC_INIT;
  float jf = 0.0f;
#pragma unroll 8
  for (int j = 0; j < NN; ++j) {
    acc = fminf(acc, fmaf(jf, ci, line[l][j])); // (f[j]+j^2) - 2*i*j
    jf += 1.0f;
  }
  dst[base + i] = acc + fi * fi;                // + i^2
}

// ---------------------------------------------------------------------------
// Passes 2/3: EDT along a strided axis. Block = (32, 8).
// Tile = 160 lines x 32 contiguous x (20KB LDS), DMA'd by the TDM with one
// tensor_load_to_lds (2D tile: tile_dim0 = 32 f32, tile_dim1 = 160,
// tensor_dim0_stride = lineStride elements). TENSORcnt tracks completion.
//   pass H: planeStride = N^2 (blockIdx.y = z), lineStride = N   (j = y)
//   pass D: planeStride = N   (blockIdx.y = y), lineStride = N^2 (j = z)
// ---------------------------------------------------------------------------
__global__ __launch_bounds__(256) void edt_pass_strided(const float* __restrict__ src,
                                                        float* __restrict__ dst,
                                                        int planeStride,
                                                        int lineStride) {
  __shared__ __align__(16) float tile[NN * 32]; // tile[j*32 + x']
  const int tx = threadIdx.x;                   // 0..31 : x within tile
  const int ty = threadIdx.y;                   // 0..7  : wave id
  const int x0 = blockIdx.x * 32;
  const size_t base = (size_t)blockIdx.z * NN3 +
                      (size_t)blockIdx.y * planeStride + x0;

  if (ty == 0) {
    // Build Tensor DMA Descriptor (D#), groups 0 and 1 (2D tile form).
    const uint64_t ga = (uint64_t)(uintptr_t)(src + base);
    const uint32_t la = lds_byte_addr(&tile[0]);
    // Group 0: count=1 | lds_addr | global_addr[56:0] | type=2 ("image")
    u32x4 g0 = { 1u,
                 la,
                 (uint32_t)ga,
                 ((uint32_t)(ga >> 32) & 0x01FFFFFFu) | (2u << 30) };
    // Group 1 (bit-packed, see ISA 8.4):
    //  s0: workgroup_mask=0, data_size=2 (4B), no barrier/iterate/pad
    //  tensor_dim0/1 = 0x7FFFFFFF (tile fully in-bounds; disable clipping)
    //  tile_dim0 = 32, tile_dim1 = 160, tile_dim2 = 0
    //  tensor_dim0_stride = lineStride (elements); dim1 stride unused
    const uint32_t S = (uint32_t)lineStride;
    u32x8 g1 = { (2u << 16),
                 0xFFFF0000u,                      // tensor_dim0[15:0] << 16
                 0xFFFF7FFFu,                      // t_dim1[15:0]<<16 | t_dim0[31:16]
                 (32u << 16) | 0x7FFFu,            // tile_dim0<<16 | t_dim1[31:16]
                 160u,                             // tile_dim2=0 <<16 | tile_dim1
                 S,                                // tensor_dim0_stride[31:0]
                 0u,                               // dim1_stride[15:0]<<16 | dim0_stride[47:32]
                 0u };
    asm volatile("tensor_load_to_lds %0, %1" :: "s"(g0), "s"(g1) : "memory");
    __builtin_amdgcn_s_wait_tensorcnt(0);
  }
  __syncthreads();

  // Fold w[j] = v[j] + j^2 into the tile once (shared by all 160 outputs).
#pragma unroll 4
  for (int j = ty; j < NN; j += 8) {
    tile[j * 32 + tx] += (float)(j * j);
  }
  __syncthreads();

  // Each thread: 20 outputs i = ty + 8k, as 2 groups of 10 accumulators.
  // Per candidate j: 1 ds_load + 1 add + 10 fma + 10 min  (2.2 VALU/output).
  for (int g = 0; g < 2; ++g) {
    float acc[10], ci[10];
#pragma unroll
    for (int m = 0; m < 10; ++m) {
      acc[m] = ACC_INIT;
      ci[m] = -2.0f * (float)(ty + 80 * g + 8 * m);
    }
    float jf = 0.0f;
#pragma unroll 4
    for (int j = 0; j < NN; ++j) {
      const float w = tile[j * 32 + tx];          // conflict-free: 32 lanes, 32 banks
#pragma unroll
      for (int m = 0; m < 10; ++m) {
        acc[m] = fminf(acc[m], fmaf(jf, ci[m], w));
      }
      jf += 1.0f;
    }
#pragma unroll
    for (int m = 0; m < 10; ++m) {
      const int i = ty + 80 * g + 8 * m;
      const float fi = (float)i;
      dst[base + (size_t)i * lineStride + tx] = acc[m] + fi * fi;
    }
  }
}

// ---------------------------------------------------------------------------
// Final: ridge detect on d2 (sqrt is monotone -> compare d2 directly) * img.
// Clamped 27-neighbor reads are exact for the -inf-padded 'SAME' window
// (clamped duplicates are still in-window values). L2-resident traffic.
// ---------------------------------------------------------------------------
__global__ __launch_bounds__(256) void skel_final(const float* __restrict__ d2,
                                                  const float* __restrict__ img,
                                                  float* __restrict__ out) {
  const int idx = blockIdx.x * 256 + threadIdx.x;
  if (idx >= 4 * NN3) return;
  int x = idx % NN;
  int t = idx / NN;
  int y = t % NN;  t /= NN;
  int z = t % NN;
  const int b = t / NN;
  const float* vb = d2 + (size_t)b * NN3;

  const float c = vb[(z * NN + y) * NN + x];
  float mx = -1.0f;
#pragma unroll
  for (int dz = -1; dz <= 1; ++dz) {
    const int zz = min(max(z + dz, 0), NN - 1);
#pragma unroll
    for (int dy = -1; dy <= 1; ++dy) {
      const int yy = min(max(y + dy, 0), NN - 1);
#pragma unroll
      for (int dx = -1; dx <= 1; ++dx) {
        const int xx = min(max(x + dx, 0), NN - 1);
        mx = fmaxf(mx, vb[(zz * NN + yy) * NN + xx]);
      }
    }
  }
  const float s = ((c >= mx) && (c > 0.0f)) ? 1.0f : 0.0f;
  out[idx] = s * img[idx];
}

// ---------------------------------------------------------------------------
extern "C" void kernel_launch(void* const* d_in, const int* in_sizes, int n_in,
                              void* d_out, int out_size, void* d_ws, size_t ws_size,
                              hipStream_t stream) {
  (void)in_sizes; (void)n_in; (void)out_size; (void)ws_size;
  const float* img = (const float*)d_in[0];     // (4,1,160,160,160) f32
  float* out = (float*)d_out;                   // same shape
  float* ws  = (float*)d_ws;                    // needs 4*160^3 f32 = 65.5MB
  float* tmp = out;                             // out doubles as pass-2 scratch

  dim3 b1(NN, 2), g1(NN / 2, NN, 4);
  edt_pass_w<<<g1, b1, 0, stream>>>(img, ws);                    // along W -> ws

  dim3 b2(32, 8), g2(NN / 32, NN, 4);
  edt_pass_strided<<<g2, b2, 0, stream>>>(ws, tmp, NN2, NN);     // along H -> out(scratch)
  edt_pass_strided<<<g2, b2, 0, stream>>>(tmp, ws, NN, NN2);     // along D -> ws

  skel_final<<<(4 * NN3 + 255) / 256, 256, 0, stream>>>(ws, img, out);
}